// ProductManifoldAttention_43052752175140
// MI455X (gfx1250) — compile-verified
//
#include <hip/hip_runtime.h>
#include <math.h>

typedef float v2f __attribute__((ext_vector_type(2)));
typedef float v4f __attribute__((ext_vector_type(4)));
typedef float v8f __attribute__((ext_vector_type(8)));
typedef int   v4i __attribute__((ext_vector_type(4)));

#define BATCH   4
#define S_LEN   2048
#define DIM     1024
#define NH      4
#define HD      256

// ---- CDNA5 async global->LDS staging (ASYNCcnt path), guarded ------------
#if defined(__has_builtin)
#  if __has_builtin(__builtin_amdgcn_global_load_async_to_lds_b128) && \
      __has_builtin(__builtin_amdgcn_s_wait_asynccnt)
#    define USE_ASYNC_LDS 1
#  endif
#endif
#ifndef USE_ASYNC_LDS
#  define USE_ASYNC_LDS 0
#endif

__device__ __forceinline__ void stage16(const float* __restrict__ g,
                                        float* __restrict__ l) {
#if USE_ASYNC_LDS
  __builtin_amdgcn_global_load_async_to_lds_b128(
      (__attribute__((address_space(1))) v4i*)g,
      (__attribute__((address_space(3))) v4i*)l, 0, 0);
#else
  *(v4f*)l = *(const v4f*)g;
#endif
}

__device__ __forceinline__ void stage_wait() {
#if USE_ASYNC_LDS
  __builtin_amdgcn_s_wait_asynccnt(0);
#endif
}

__device__ __forceinline__ v8f wmma_f32(v2f a, v2f b, v8f c) {
  // D = A(16x4) * B(4x16) + C(16x16), fp32 throughout.
  return __builtin_amdgcn_wmma_f32_16x16x4_f32(
      /*neg_a=*/false, a, /*neg_b=*/false, b,
      /*c_mod=*/(short)0, c, /*reuse_a=*/false, /*reuse_b=*/false);
}

// ---------------------------------------------------------------------------
// Kernel 1: head projection.  xh[b,h,s,e] = x[b,s,:] @ W_proj[h] + b_proj[h]
// Also writes xhT2[b,h,e/2,s,2] (k-pair interleaved transpose) so the Gram
// GEMM B-fragments are single coalesced b64 loads.
// grid = (HD/64, B*S/64, NH), block = 256 (8 waves)
// ---------------------------------------------------------------------------
__global__ __launch_bounds__(256) void proj_kernel(
    const float* __restrict__ x, const float* __restrict__ Wp,
    const float* __restrict__ bp, float* __restrict__ xh,
    float* __restrict__ xhT2) {
  __shared__ float Alds[64 * 32];            // row-major (async-staged)
  __shared__ float Blds[32 * 64];            // k-pair interleaved [k/2][n][2]
  const int h  = blockIdx.z;
  const int m0 = blockIdx.y * 64;
  const int n0 = blockIdx.x * 64;
  const int t    = threadIdx.x;
  const int wave = t >> 5, lane = t & 31;
  const int mn = lane & 15, half = lane >> 4;
  const int wr = wave & 3, wc = wave >> 2;   // wave tile: rows wr*16, cols wc*32
  const float* Bsrc = Wp + (size_t)h * DIM * HD;

  v8f acc0 = {}; v8f acc1 = {};
  for (int kc = 0; kc < DIM; kc += 32) {
    __syncthreads();
    // A tile 64x32 via b128 (async if available)
#pragma unroll
    for (int j = 0; j < 2; ++j) {
      int elem = (t + 256 * j) * 4;
      int r = elem >> 5, c = elem & 31;
      stage16(&x[(size_t)(m0 + r) * DIM + kc + c], &Alds[elem]);
    }
    // B tile 32x64: vector load, scatter into k-pair layout
#pragma unroll
    for (int j = 0; j < 2; ++j) {
      int elem = (t + 256 * j) * 4;
      int r = elem >> 6, c = elem & 63;
      v4f v = *(const v4f*)&Bsrc[(size_t)(kc + r) * HD + n0 + c];
#pragma unroll
      for (int u = 0; u < 4; ++u)
        Blds[(r >> 1) * 128 + (c + u) * 2 + (r & 1)] = v[u];
    }
    stage_wait();
    __syncthreads();
#pragma unroll
    for (int ks = 0; ks < 8; ++ks) {
      int klo = ks * 4 + half * 2;
      v2f a  = *(const v2f*)&Alds[(wr * 16 + mn) * 32 + klo];
      v2f b0 = *(const v2f*)&Blds[(klo >> 1) * 128 + (wc * 32 + mn) * 2];
      v2f b1 = *(const v2f*)&Blds[(klo >> 1) * 128 + (wc * 32 + 16 + mn) * 2];
      acc0 = wmma_f32(a, b0, acc0);
      acc1 = wmma_f32(a, b1, acc1);
    }
  }
  const float* bias = bp + h * HD;
#pragma unroll
  for (int r = 0; r < 8; ++r) {
    int m   = r + half * 8;
    int row = m0 + wr * 16 + m;              // row = b*S + s
    int b   = row >> 11, s = row & 2047;
    size_t hb = ((size_t)b * NH + h);
    int e0 = n0 + wc * 32 + mn;
    int e1 = e0 + 16;
    float v0 = acc0[r] + bias[e0];
    float v1 = acc1[r] + bias[e1];
    xh[(hb * S_LEN + s) * HD + e0] = v0;
    xh[(hb * S_LEN + s) * HD + e1] = v1;
    size_t tb = hb * HD * S_LEN;
    xhT2[tb + ((((size_t)(e0 >> 1)) * S_LEN + s) << 1) + (e0 & 1)] = v0;
    xhT2[tb + ((((size_t)(e1 >> 1)) * S_LEN + s) << 1) + (e1 & 1)] = v1;
  }
}

// ---------------------------------------------------------------------------
// Kernel 2: per-row manifold stats.  sq = |xh|^2 ; aux = t0 or 1/(|x|+eps)
// grid = B*NH*S/8 blocks of 256 (8 waves = 8 rows)
// ---------------------------------------------------------------------------
__global__ __launch_bounds__(256) void stats_kernel(
    const float* __restrict__ xh, float* __restrict__ stats) {
  const int wave = threadIdx.x >> 5, lane = threadIdx.x & 31;
  const int row = blockIdx.x * 8 + wave;     // row in [0, B*NH*S)
  const float* p = xh + (size_t)row * HD;
  float sq = 0.f;
#pragma unroll
  for (int i = lane; i < HD; i += 32) { float v = p[i]; sq += v * v; }
  for (int off = 16; off; off >>= 1) sq += __shfl_xor(sq, off, 32);
  if (lane == 0) {
    int h = (row >> 11) & 3;                 // head index
    float aux;
    if (h == 1)      aux = 1.0f / (sqrtf(sq) + 1e-8f);   // spherical inv-norm
    else if (h == 2) aux = 0.0f;                          // euclidean unused
    else             aux = sqrtf(1.0f + sq + 1e-8f);      // hyperbolic t0
    stats[row * 2 + 0] = sq;
    stats[row * 2 + 1] = aux;
  }
}

// ---------------------------------------------------------------------------
// Kernel 3: fused attention for one (b,h) x 16-row strip.
// 320KB CDNA5 LDS: keep the whole 16x2048 f32 score strip resident.
//   phase 1: Gram via WMMA  -> manifold distance -> scaled scores into LDS
//   phase 2: row softmax in LDS (wave32 shuffle reductions), write attn out
//   phase 3: out = P @ xh via WMMA, write head-slice of `combined`
// grid = (B*NH, S/16), block = 256 (8 waves)
// ---------------------------------------------------------------------------
__global__ __launch_bounds__(256) void attn_kernel(
    const float* __restrict__ xh, const float* __restrict__ xhT2,
    const float* __restrict__ stats, float* __restrict__ attn_out,
    float* __restrict__ combined) {
  __shared__ float sc[16 * S_LEN];           // 128 KB score strip
  __shared__ float Arow[16 * HD];            // 16 KB row block of xh
  __shared__ float rowsq[16], rowaux[16];

  const int bh = blockIdx.x;                 // b*NH + h
  const int h = bh & 3, btch = bh >> 2;
  const int htype = (h == 1) ? 1 : ((h == 2) ? 2 : 0);
  const int r0 = blockIdx.y * 16;
  const int t = threadIdx.x, wave = t >> 5, lane = t & 31;
  const int mn = lane & 15, half = lane >> 4;
  const size_t xhBase  = (size_t)bh * S_LEN * HD;
  const size_t xhTBase = (size_t)bh * HD * S_LEN;

  // stage A rows (async b128 if available): 16x256 floats
#pragma unroll
  for (int j = 0; j < 4; ++j) {
    int elem = (t + 256 * j) * 4;
    int row = elem >> 8, col = elem & 255;
    stage16(&xh[xhBase + (size_t)(r0 + row) * HD + col], &Arow[elem]);
  }
  if (t < 16)       rowsq[t]       = stats[(bh * S_LEN + r0 + t) * 2];
  else if (t < 32)  rowaux[t - 16] = stats[(bh * S_LEN + r0 + t - 16) * 2 + 1];
  stage_wait();
  __syncthreads();

  // ---- phase 1: scores -------------------------------------------------
  const float scale = 0.0625f;               // 1/sqrt(256)
  for (int ct = 0; ct < 16; ++ct) {
    int c = wave * 256 + ct * 16 + mn;       // this lane's column
    float csq  = stats[(bh * S_LEN + c) * 2];
    float caux = stats[(bh * S_LEN + c) * 2 + 1];
    v8f acc = {};
#pragma unroll 4
    for (int ks = 0; ks < 64; ++ks) {
      int klo = ks * 4 + half * 2;           // even
      v2f a = *(const v2f*)&Arow[mn * HD + klo];
      v2f b = *(const v2f*)&xhT2[xhTBase +
                                 ((((size_t)(klo >> 1)) * S_LEN + c) << 1)];
      acc = wmma_f32(a, b, acc);
    }
#pragma unroll
    for (int r = 0; r < 8; ++r) {
      int m = r + half * 8;
      float g = acc[r];
      float d;
      if (htype == 0) {                      // hyperbolic (Lorentz)
        float inner = g - rowaux[m] * caux;
        float z = fmaxf(-inner, 1.0f + 1e-7f);
        d = acoshf(z);
      } else if (htype == 1) {               // spherical
        float cs = g * rowaux[m] * caux;
        cs = fminf(fmaxf(cs, -1.0f + 1e-7f), 1.0f - 1e-7f);
        d = acosf(cs);
      } else {                               // euclidean
        float d2 = rowsq[m] + csq - 2.0f * g;
        d = sqrtf(fmaxf(d2, 1e-8f));
      }
      sc[m * S_LEN + c] = -d * scale;
    }
  }
  __syncthreads();

  // ---- phase 2: softmax + write attention ------------------------------
  for (int rr = wave; rr < 16; rr += 8) {
    float mx = -3.4e38f;
    for (int c = lane; c < S_LEN; c += 32) mx = fmaxf(mx, sc[rr * S_LEN + c]);
    for (int off = 16; off; off >>= 1) mx = fmaxf(mx, __shfl_xor(mx, off, 32));
    float sum = 0.f;
    for (int c = lane; c < S_LEN; c += 32) {
      float e = __expf(sc[rr * S_LEN + c] - mx);
      sc[rr * S_LEN + c] = e;
      sum += e;
    }
    for (int off = 16; off; off >>= 1) sum += __shfl_xor(sum, off, 32);
    float inv = 1.0f / sum;
    size_t aBase = ((size_t)bh * S_LEN + (r0 + rr)) * S_LEN;
    for (int c = lane; c < S_LEN; c += 32) {
      float p = sc[rr * S_LEN + c] * inv;
      sc[rr * S_LEN + c] = p;
      attn_out[aBase + c] = p;               // coalesced 268MB stream
    }
  }
  __syncthreads();

  // ---- phase 3: out = P(16x2048) @ xh(2048x256) ------------------------
  v8f o0 = {}, o1 = {};
  const int e0 = wave * 32;
#pragma unroll 4
  for (int k = 0; k < S_LEN; k += 4) {
    int klo = k + half * 2;
    v2f a = *(const v2f*)&sc[mn * S_LEN + klo];
    const float* xp = &xh[xhBase + (size_t)klo * HD + e0 + mn];
    v2f b0, b1;
    b0.x = xp[0];   b0.y = xp[HD];
    b1.x = xp[16];  b1.y = xp[HD + 16];
    o0 = wmma_f32(a, b0, o0);
    o1 = wmma_f32(a, b1, o1);
  }
#pragma unroll
  for (int r = 0; r < 8; ++r) {
    int m = r + half * 8;
    int s = r0 + m;
    size_t base = ((size_t)btch * S_LEN + s) * DIM + h * HD + e0 + mn;
    combined[base]      = o0[r];
    combined[base + 16] = o1[r];
  }
}

// ---------------------------------------------------------------------------
// Kernel 4: output projection.  out = combined @ W_out + b_out
// grid = (DIM/64, B*S/64), block = 256
// ---------------------------------------------------------------------------
__global__ __launch_bounds__(256) void outproj_kernel(
    const float* __restrict__ combined, const float* __restrict__ Wo,
    const float* __restrict__ bo, float* __restrict__ out) {
  __shared__ float Alds[64 * 32];            // row-major (async-staged)
  __shared__ float Blds[32 * 64];            // k-pair interleaved [k/2][n][2]
  const int m0 = blockIdx.y * 64;
  const int n0 = blockIdx.x * 64;
  const int t = threadIdx.x;
  const int wave = t >> 5, lane = t & 31;
  const int mn = lane & 15, half = lane >> 4;
  const int wr = wave & 3, wc = wave >> 2;

  v8f acc0 = {}; v8f acc1 = {};
  for (int kc = 0; kc < DIM; kc += 32) {
    __syncthreads();
#pragma unroll
    for (int j = 0; j < 2; ++j) {
      int elem = (t + 256 * j) * 4;
      int r = elem >> 5, c = elem & 31;
      stage16(&combined[(size_t)(m0 + r) * DIM + kc + c], &Alds[elem]);
    }
#pragma unroll
    for (int j = 0; j < 2; ++j) {
      int elem = (t + 256 * j) * 4;
      int r = elem >> 6, c = elem & 63;
      v4f v = *(const v4f*)&Wo[(size_t)(kc + r) * DIM + n0 + c];
#pragma unroll
      for (int u = 0; u < 4; ++u)
        Blds[(r >> 1) * 128 + (c + u) * 2 + (r & 1)] = v[u];
    }
    stage_wait();
    __syncthreads();
#pragma unroll
    for (int ks = 0; ks < 8; ++ks) {
      int klo = ks * 4 + half * 2;
      v2f a  = *(const v2f*)&Alds[(wr * 16 + mn) * 32 + klo];
      v2f b0 = *(const v2f*)&Blds[(klo >> 1) * 128 + (wc * 32 + mn) * 2];
      v2f b1 = *(const v2f*)&Blds[(klo >> 1) * 128 + (wc * 32 + 16 + mn) * 2];
      acc0 = wmma_f32(a, b0, acc0);
      acc1 = wmma_f32(a, b1, acc1);
    }
  }
#pragma unroll
  for (int r = 0; r < 8; ++r) {
    int m   = r + half * 8;
    int row = m0 + wr * 16 + m;
    int e0  = n0 + wc * 32 + mn;
    out[(size_t)row * DIM + e0]      = acc0[r] + bo[e0];
    out[(size_t)row * DIM + e0 + 16] = acc1[r] + bo[e0 + 16];
  }
}

// ---------------------------------------------------------------------------
extern "C" void kernel_launch(void* const* d_in, const int* in_sizes, int n_in,
                              void* d_out, int out_size, void* d_ws,
                              size_t ws_size, hipStream_t stream) {
  (void)in_sizes; (void)n_in; (void)out_size; (void)ws_size;
  const float* x  = (const float*)d_in[0];
  const float* Wp = (const float*)d_in[1];
  const float* bp = (const float*)d_in[2];
  const float* Wo = (const float*)d_in[3];
  const float* bo = (const float*)d_in[4];

  float* out  = (float*)d_out;                          // [B,S,DIM]
  float* attn = out + (size_t)BATCH * S_LEN * DIM;      // [B,H,S,S]

  float* ws       = (float*)d_ws;
  float* xh       = ws;                                             // [B,H,S,HD]
  float* xhT2     = xh    + (size_t)BATCH * NH * S_LEN * HD;        // [B,H,HD/2,S,2]
  float* stats    = xhT2  + (size_t)BATCH * NH * S_LEN * HD;        // [B,H,S,2]
  float* combined = stats + (size_t)BATCH * NH * S_LEN * 2;         // [B,S,DIM]

  dim3 g1(HD / 64, (BATCH * S_LEN) / 64, NH);           // (4,128,4)
  proj_kernel<<<g1, 256, 0, stream>>>(x, Wp, bp, xh, xhT2);

  stats_kernel<<<(BATCH * NH * S_LEN) / 8, 256, 0, stream>>>(xh, stats);

  dim3 g2(BATCH * NH, S_LEN / 16);                      // (16,128)
  attn_kernel<<<g2, 256, 0, stream>>>(xh, xhT2, stats, attn, combined);

  dim3 g3(DIM / 64, (BATCH * S_LEN) / 64);              // (16,128)
  outproj_kernel<<<g3, 256, 0, stream>>>(combined, Wo, bo, out);
}